// GroupConvLayer_41008347742649
// MI455X (gfx1250) — compile-verified
//
#include <hip/hip_runtime.h>
#include <hip/hip_bf16.h>

typedef __attribute__((ext_vector_type(2))) float v2f;
typedef __attribute__((ext_vector_type(8))) float v8f;

#define CDIM 64
#define PDIM 16
#define KSPL 9
#define NCOL 144   // P*K = 16*9
#define INV2S 0.35355339059327373f  // 1/(2*sqrt(2))

// ---------------------------------------------------------------------------
// Kernel 0: out[n*16+p] = bias[p]   (fresh every launch; edge kernel accumulates)
// ---------------------------------------------------------------------------
__global__ void init_out_kernel(float* __restrict__ out,
                                const float* __restrict__ bias, int total) {
    int i = blockIdx.x * blockDim.x + threadIdx.x;
    if (i < total) out[i] = bias[i & (PDIM - 1)];
}

// ---------------------------------------------------------------------------
// Kernel 1: Z[n, p*9+k] = sum_c x[n,c] * W[p,k,c]
// GEMM: A = x (N x 64), B = Wf^T (64 x 144), Wf = W flattened (144 x 64).
// One wave per 16x16 tile of Z using V_WMMA_F32_16X16X4_F32 (fp32-exact).
//
// A 16x4 f32 fragment (ISA 7.12.2): lanes 0-15 row M=lane hold K=0(v0),K=1(v1);
// lanes 16-31 row M=lane-16 hold K=2(v0),K=3(v1).
// B 4x16 fragment row-striped: v0 = rows K=0 (lanes 0-15) / K=2 (lanes 16-31),
// v1 = rows K=1 / K=3 (mirrors C/D striping).
// ---------------------------------------------------------------------------
__global__ void zgemm_wmma_kernel(const float* __restrict__ x,
                                  const float* __restrict__ Wf,
                                  float* __restrict__ Z,
                                  int N, int ntilesM) {
    int wave = (blockIdx.x * blockDim.x + threadIdx.x) >> 5;
    int lane = threadIdx.x & 31;
    int totalTiles = ntilesM * (NCOL / 16);
    if (wave >= totalTiles) return;

    int mtile = wave / (NCOL / 16);
    int ntile = wave % (NCOL / 16);
    int rowBase = mtile * 16;

    int mrow = rowBase + (lane & 15);
    int arow = (mrow < N) ? mrow : (N - 1);          // clamp, keep EXEC all-1s
    int koff = (lane >> 4) << 1;                     // 0 for lanes 0-15, 2 for 16-31
    int ncol = ntile * 16 + (lane & 15);

    const float* __restrict__ xr = x  + (size_t)arow * CDIM;
    const float* __restrict__ wr = Wf + (size_t)ncol * CDIM;

    v8f acc = {};
#pragma unroll
    for (int k = 0; k < CDIM; k += 4) {
        v2f a, b;
        a.x = xr[k + koff];
        a.y = xr[k + koff + 1];
        b.x = wr[k + koff];
        b.y = wr[k + koff + 1];
        // (neg_a, A, neg_b, B, c_mod, C, reuse_a, reuse_b)
        acc = __builtin_amdgcn_wmma_f32_16x16x4_f32(
            false, a, false, b, (short)0, acc, false, false);
    }

    // C/D layout: VGPR r -> M=r (lanes 0-15) / M=r+8 (lanes 16-31), N = lane&15
    int nOut = ntile * 16 + (lane & 15);
    int mHi  = (lane >> 4) * 8;
#pragma unroll
    for (int r = 0; r < 8; ++r) {
        int M = rowBase + r + mHi;
        if (M < N) Z[(size_t)M * NCOL + nOut] = acc[r];
    }
}

// ---------------------------------------------------------------------------
// Kernel 2: per (edge, p) message + scatter-add.
// 16 consecutive lanes share one edge -> pose read is a 128B contiguous burst,
// pos reads broadcast, Z gathers land in a 576B window of L2-resident Z[col].
// ---------------------------------------------------------------------------
__global__ void edge_kernel(const long long* __restrict__ ei,   // (2,E)
                            const float* __restrict__ pos,      // (N,2)
                            const float* __restrict__ pose,     // (N,16,2)
                            const float* __restrict__ Z,        // (N,144)
                            float* __restrict__ out,            // (N,16)
                            int E) {
    long long idx = (long long)blockIdx.x * blockDim.x + threadIdx.x;
    if (idx >= (long long)E * PDIM) return;
    int e = (int)(idx >> 4);
    int p = (int)(idx & (PDIM - 1));

    long long r  = ei[e];                 // row (destination)
    long long cN = ei[(size_t)E + e];     // col (source)

    float q0 = (pos[2 * cN]     - pos[2 * r])     * INV2S;
    float q1 = (pos[2 * cN + 1] - pos[2 * r + 1]) * INV2S;

    const float* ps = pose + ((size_t)r * PDIM + p) * 2;
    float a = ps[0], b = ps[1];

    // p = R * q + 0.5, R = [[a,-b],[b,a]]
    float p0 = fmaf(a, q0, fmaf(-b, q1, 0.5f));
    float p1 = fmaf(b, q0, fmaf( a, q1, 0.5f));

    // degree-1 open B-spline over k=3 grid (exact reference semantics)
    float v0 = fminf(fmaxf(p0, 0.0f), 1.0f) * 2.0f;
    float v1 = fminf(fmaxf(p1, 0.0f), 1.0f) * 2.0f;
    float lo0 = floorf(v0), lo1 = floorf(v1);
    float f0 = v0 - lo0,    f1 = v1 - lo1;
    int l0 = (int)lo0, l1 = (int)lo1;
    int i00 = min(max(l0,     0), 2);
    int i01 = min(max(l0 + 1, 0), 2);
    int i10 = min(max(l1,     0), 2);
    int i11 = min(max(l1 + 1, 0), 2);

    float g0 = 1.0f - f0, g1 = 1.0f - f1;

    const float* __restrict__ Zc = Z + (size_t)cN * NCOL + p * KSPL;
    float m = g0 * g1 * Zc[i00 + 3 * i10]
            + g0 * f1 * Zc[i00 + 3 * i11]
            + f0 * g1 * Zc[i01 + 3 * i10]
            + f0 * f1 * Zc[i01 + 3 * i11];

    atomicAdd(out + (size_t)r * PDIM + p, m);
}

// ---------------------------------------------------------------------------
extern "C" void kernel_launch(void* const* d_in, const int* in_sizes, int n_in,
                              void* d_out, int out_size, void* d_ws, size_t ws_size,
                              hipStream_t stream) {
    const float*     x    = (const float*)d_in[0];     // (N,64)
    const float*     pos  = (const float*)d_in[1];     // (N,2)
    const float*     pose = (const float*)d_in[2];     // (N,16,2)
    const float*     Wf   = (const float*)d_in[3];     // (16,9,64) -> (144,64)
    const float*     bias = (const float*)d_in[4];     // (16,)
    const long long* ei   = (const long long*)d_in[5]; // (2,E) int64

    int N = in_sizes[1] / 2;
    int E = in_sizes[5] / 2;

    float* Z   = (float*)d_ws;     // N*144 floats (~28.8 MB), L2-resident
    float* out = (float*)d_out;    // (N,16)

    // 0) out = bias broadcast
    int totalOut = N * PDIM;
    init_out_kernel<<<(totalOut + 255) / 256, 256, 0, stream>>>(out, bias, totalOut);

    // 1) Z = x @ Wf^T  via fp32 WMMA
    int ntilesM = (N + 15) / 16;
    int waves   = ntilesM * (NCOL / 16);
    int threads = waves * 32;
    zgemm_wmma_kernel<<<(threads + 255) / 256, 256, 0, stream>>>(x, Wf, Z, N, ntilesM);

    // 2) edge messages + scatter-add
    long long tot = (long long)E * PDIM;
    int blocks = (int)((tot + 255) / 256);
    edge_kernel<<<blocks, 256, 0, stream>>>(ei, pos, pose, Z, out, E);
}